// NerfRenderer_17961553232045
// MI455X (gfx1250) — compile-verified
//
#include <hip/hip_runtime.h>
#include <hip/hip_fp16.h>
#include <math.h>

// NeRF log-depth volume rendering for MI455X (gfx1250, wave32).
// Bandwidth-bound (~907 MB @ 23.3 TB/s => ~39us floor). One wave32 per ray;
// cumprod done as a 5-step shuffle product-scan with a scalar carry.
// sigma+rgb streams are double-buffered into LDS with the gfx1250 async
// copy engine (GLOBAL_LOAD_ASYNC_TO_LDS_B128 / s_wait_asynccnt).

#define S_TOTAL 576
#define S_INNER 384
#define NCHUNK  (S_TOTAL / 32)   // 18 chunks of 32 samples

#if __has_builtin(__builtin_amdgcn_global_load_async_to_lds_b128)
#define HAVE_ASYNC_COPY 1
typedef int v4i __attribute__((vector_size(4 * sizeof(int))));
typedef __attribute__((address_space(1))) v4i as1_v4i;   // global int4
typedef __attribute__((address_space(3))) v4i as3_v4i;   // LDS int4
#else
#define HAVE_ASYNC_COPY 0
#endif

// ---------------------------------------------------------------------------
// Kernel 1: per-sample constant tables (same for every ray), computed in fp64.
//   ws[0..575]        = z_vals_log  (fp32, pre-fp16 — this is the output grid)
//   ws[576..1151]     = z           (fp32 value of fp16(10^zlog), RNE)
//   ws[1152..1727]    = delta       (z[s+1]-z[s], last = 1e10)
// ---------------------------------------------------------------------------
__device__ __forceinline__ double zlog_of(int i) {
    const double L0      = -1.3010299956639811952137389;   // log10(0.05)
    const double STEP_IN = (0.0 - L0) / 384.0;             // (b-a)/steps
    const double STEP_OUT = 2.0 / 192.0;                   // log10(100)/192
    return (i < S_INNER) ? (L0 + (double)i * STEP_IN)
                         : ((double)(i - S_INNER) * STEP_OUT);
}

__global__ void nerf_tables_kernel(float* __restrict__ ws) {
    int s = blockIdx.x * blockDim.x + threadIdx.x;
    if (s >= S_TOTAL) return;
    double zl = zlog_of(s);
    float zq = __half2float(__float2half((float)exp10(zl)));   // fp16 RNE quantize
    float delta;
    if (s == S_TOTAL - 1) {
        delta = 1e10f;                                          // FAR_DELTA
    } else {
        float zq1 = __half2float(__float2half((float)exp10(zlog_of(s + 1))));
        delta = zq1 - zq;                                       // fp32 sub of quantized z
    }
    ws[s]               = (float)zl;
    ws[S_TOTAL + s]     = zq;
    ws[2 * S_TOTAL + s] = delta;
}

// ---------------------------------------------------------------------------
// Kernel 2: one wave32 per ray; 8 rays per 256-thread block.
// Per chunk (32 samples): 128 B sigma + 384 B rgb = 512 B staged to LDS by a
// single lane-partitioned b128 async copy (lanes 0-7 sigma, 8-31 rgb).
// ---------------------------------------------------------------------------
__global__ __launch_bounds__(256)
void nerf_render_kernel(const float* __restrict__ sigmas,   // [N,576]
                        const float* __restrict__ rgbs,     // [N,576,3]
                        const float* __restrict__ bg,       // [3]
                        const float* __restrict__ tables,   // ws
                        float* __restrict__ out_image,      // [N,3]
                        float* __restrict__ out_invd,       // [N]
                        float* __restrict__ out_w,          // [N,576]
                        float* __restrict__ out_zlog,       // [N,576]
                        int n_rays) {
    __shared__ float s_zlog[S_TOTAL];
    __shared__ float s_zq[S_TOTAL];
    __shared__ float s_delta[S_TOTAL];
#if HAVE_ASYNC_COPY
    __shared__ float s_pipe[8][2][128];     // per-wave double buffer: 32 sigma + 96 rgb
#endif
    for (int i = threadIdx.x; i < S_TOTAL; i += blockDim.x) {
        s_zlog[i]  = tables[i];
        s_zq[i]    = tables[S_TOTAL + i];
        s_delta[i] = tables[2 * S_TOTAL + i];
    }
    __syncthreads();

    const int lane = threadIdx.x & 31;
    const int wave = threadIdx.x >> 5;
    const int r = blockIdx.x * 8 + wave;
    if (r >= n_rays) return;                 // whole-wave uniform

    const float* sig  = sigmas + (size_t)r * S_TOTAL;       // row 16B-aligned (2304 B stride)
    const float* rgb  = rgbs   + (size_t)r * S_TOTAL * 3;   // row 16B-aligned (6912 B stride)
    float* wrow = out_w    + (size_t)r * S_TOTAL;
    float* zrow = out_zlog + (size_t)r * S_TOTAL;

    float T = 1.0f;                          // carried transmittance
    float img0 = 0.f, img1 = 0.f, img2 = 0.f, invd = 0.f, wsum = 0.f;

#if HAVE_ASYNC_COPY
    // One async b128 per lane: lanes 0-7 cover sigma[c*32..+31] (16B each),
    // lanes 8-31 cover rgb[c*96..+95]. All addresses 16B-aligned.
    auto issue_chunk = [&](int c, int buf) {
        const float* src = (lane < 8)
            ? (sig + (size_t)c * 32 + (size_t)lane * 4)
            : (rgb + (size_t)c * 96 + (size_t)(lane - 8) * 4);
        float* dst = &s_pipe[wave][buf][(lane < 8) ? (lane * 4)
                                                   : (32 + (lane - 8) * 4)];
        __builtin_amdgcn_global_load_async_to_lds_b128(
            (as1_v4i*)src, (as3_v4i*)dst, 0, 0);
    };
    issue_chunk(0, 0);
#endif

    for (int c = 0; c < NCHUNK; ++c) {
#if HAVE_ASYNC_COPY
        if (c + 1 < NCHUNK) {
            issue_chunk(c + 1, (c + 1) & 1);
            asm volatile("s_wait_asynccnt 0x1" ::: "memory");  // chunk c landed
        } else {
            asm volatile("s_wait_asynccnt 0x0" ::: "memory");
        }
        const float* buf = s_pipe[wave][c & 1];
        const int s = c * 32 + lane;
        const float sigma = buf[lane];
        const float cx = buf[32 + 3 * lane + 0];
        const float cy = buf[32 + 3 * lane + 1];
        const float cz = buf[32 + 3 * lane + 2];
#else
        const int s = c * 32 + lane;
        if (c + 1 < NCHUNK) {                 // global_prefetch_b8
            __builtin_prefetch(sig + s + 32, 0, 0);
            __builtin_prefetch(rgb + (size_t)(s + 32) * 3, 0, 0);
        }
        const float sigma = __builtin_nontemporal_load(sig + s);
        const float3 col  = *reinterpret_cast<const float3*>(rgb + (size_t)s * 3);
        const float cx = col.x, cy = col.y, cz = col.z;
#endif

        const float delta = s_delta[s];
        const float e     = expf(-sigma * delta);       // fp32, matches jnp.exp
        const float alpha = 1.0f - e;
        float p = (1.0f - alpha) + 1e-10f;              // cp_j, exact ref op order

        // Inclusive product-scan across the wave (Hillis-Steele, 5 steps).
        #pragma unroll
        for (int off = 1; off < 32; off <<= 1) {
            float o = __shfl_up(p, (unsigned)off, 32);
            if (lane >= off) p *= o;
        }
        // Exclusive prefix for this lane.
        float excl = __shfl_up(p, 1u, 32);
        if (lane == 0) excl = 1.0f;

        const float w = alpha * (T * excl);             // weights

        __builtin_nontemporal_store(w, wrow + s);       // write-once streams
        __builtin_nontemporal_store(s_zlog[s], zrow + s);

        img0 = fmaf(w, cx, img0);
        img1 = fmaf(w, cy, img1);
        img2 = fmaf(w, cz, img2);
        invd += w / s_zq[s];
        wsum += w;

        T *= __shfl(p, 31, 32);                         // fold chunk product
    }

    // Cross-lane butterfly reductions (wave32).
    #pragma unroll
    for (int off = 16; off > 0; off >>= 1) {
        img0 += __shfl_xor(img0, off, 32);
        img1 += __shfl_xor(img1, off, 32);
        img2 += __shfl_xor(img2, off, 32);
        invd += __shfl_xor(invd, off, 32);
        wsum += __shfl_xor(wsum, off, 32);
    }

    if (lane == 0) {
        const float bgw = 1.0f - wsum;
        out_image[(size_t)r * 3 + 0] = fmaf(bgw, bg[0], img0);
        out_image[(size_t)r * 3 + 1] = fmaf(bgw, bg[1], img1);
        out_image[(size_t)r * 3 + 2] = fmaf(bgw, bg[2], img2);
        out_invd[r] = invd;
    }
}

// ---------------------------------------------------------------------------
// Launch. Inputs: 0=rays_o (unused), 1=rays_d (unused), 2=bg_color[3],
// 3=sigmas[N,576], 4=rgbs[N,576,3].
// Output: image[N,3] | invdepth[N] | weights[N,576] | z_vals_log[N,576].
// ---------------------------------------------------------------------------
extern "C" void kernel_launch(void* const* d_in, const int* in_sizes, int n_in,
                              void* d_out, int out_size, void* d_ws, size_t ws_size,
                              hipStream_t stream) {
    const float* bg     = (const float*)d_in[2];
    const float* sigmas = (const float*)d_in[3];
    const float* rgbs   = (const float*)d_in[4];
    const int N = in_sizes[0] / 3;           // rays_o is [N,3]

    float* ws  = (float*)d_ws;               // 3*576 floats = 6912 B
    float* out = (float*)d_out;
    float* out_image = out;                                  // N*3
    float* out_invd  = out + (size_t)N * 3;                  // N
    float* out_w     = out + (size_t)N * 4;                  // N*576
    float* out_zlog  = out_w + (size_t)N * S_TOTAL;          // N*576

    nerf_tables_kernel<<<(S_TOTAL + 191) / 192, 192, 0, stream>>>(ws);

    const int blocks = (N + 7) / 8;          // 8 rays (waves) per block
    nerf_render_kernel<<<blocks, 256, 0, stream>>>(
        sigmas, rgbs, bg, ws, out_image, out_invd, out_w, out_zlog, N);
}